// HungarianMatcher_32384053412636
// MI455X (gfx1250) — compile-verified
//
#include <hip/hip_runtime.h>
#include <hip/hip_bf16.h>

// ---- problem constants (fixed by setup_inputs) ----
#define BQ        16000     // B*Q rows
#define NCLS      91        // classes
#define CP        92        // classes padded to multiple of 4 (WMMA K=4)
#define MTGT      2048      // targets
#define ROW_TILE  16
#define M_TILE    128       // 8 waves x 16 columns

typedef __attribute__((ext_vector_type(2))) float v2f;
typedef __attribute__((ext_vector_type(4))) float v4f;
typedef __attribute__((ext_vector_type(8))) float v8f;
typedef __attribute__((ext_vector_type(4))) int   v4i;

#define AS1 __attribute__((address_space(1)))
#define AS3 __attribute__((address_space(3)))

// typed address-space casts for the async global->LDS builtins
#define GP128(p) ((AS1 v4i*)(void*)(p))
#define LP128(p) ((AS3 v4i*)(void*)(p))
#define GP32(p)  ((AS1 int*)(void*)(p))
#define LP32(p)  ((AS3 int*)(void*)(p))

#if defined(__AMDGCN__) && __has_builtin(__builtin_amdgcn_global_load_async_to_lds_b128) && \
    __has_builtin(__builtin_amdgcn_global_load_async_to_lds_b32) && \
    __has_builtin(__builtin_amdgcn_s_wait_asynccnt)
#define USE_ASYNC 1
#else
#define USE_ASYNC 0
#endif

// ---------------------------------------------------------------------------
// Kernel 1: per-row softmax (negated, padded to CP) + cxcywh -> xyxy.
// One wave32 per row; C=91 -> lanes cover [lane, lane+32, lane+64].
// ---------------------------------------------------------------------------
__global__ __launch_bounds__(256) void matcher_prep(
    const float* __restrict__ logits,   // [BQ, 91]
    const float* __restrict__ boxes,    // [BQ, 4] cxcywh
    float* __restrict__ negprob,        // [BQ, CP] = -softmax, col 91 = 0
    float* __restrict__ xyxy) {         // [BQ, 4]
  const int wave = (blockIdx.x * blockDim.x + threadIdx.x) >> 5;
  const int lane = threadIdx.x & 31;
  if (wave >= BQ) return;

  const float* lrow = logits + (size_t)wave * NCLS;
  float x0 = lrow[lane];
  float x1 = lrow[lane + 32];
  float x2 = (lane < NCLS - 64) ? lrow[lane + 64] : -__builtin_inff();

  float m = fmaxf(x0, fmaxf(x1, x2));
  #pragma unroll
  for (int off = 16; off; off >>= 1) m = fmaxf(m, __shfl_xor(m, off, 32));

  float e0 = __expf(x0 - m);
  float e1 = __expf(x1 - m);
  float e2 = (lane < NCLS - 64) ? __expf(x2 - m) : 0.0f;
  float s = e0 + e1 + e2;
  #pragma unroll
  for (int off = 16; off; off >>= 1) s += __shfl_xor(s, off, 32);

  const float ninv = -1.0f / s;
  float* orow = negprob + (size_t)wave * CP;
  orow[lane]      = e0 * ninv;
  orow[lane + 32] = e1 * ninv;
  if (lane + 64 < CP)  // lane < 28: lane 27 writes the zero pad column 91
    orow[lane + 64] = (lane < NCLS - 64) ? e2 * ninv : 0.0f;

  if (lane < 4) {
    const float c  = boxes[(size_t)wave * 4 + (lane & 1)];
    const float wh = boxes[(size_t)wave * 4 + (lane & 1) + 2];
    xyxy[(size_t)wave * 4 + lane] = c + ((lane < 2) ? -0.5f : 0.5f) * wh;
  }
}

// ---------------------------------------------------------------------------
// Kernel 2: 16-row x 128-target tiles. cost_class via V_WMMA_F32_16X16X4_F32
// (negprob tile x one-hot(tgt_ids)), L1 + GIoU fused in the epilogue.
// Tile operands staged through LDS with async global->LDS loads.
// ---------------------------------------------------------------------------
__global__ __launch_bounds__(256) void matcher_cost(
    const float* __restrict__ negprob,  // [BQ, CP]
    const float* __restrict__ xyxy,     // [BQ, 4]
    const float* __restrict__ tboxes,   // [M, 4] xyxy
    const int*   __restrict__ tids,     // [M]
    float* __restrict__ out) {          // [BQ, M]
  __shared__ __align__(16) float s_prob[ROW_TILE * CP];  // 16 x 92 (stride 92: bank-conflict free)
  __shared__ __align__(16) float s_rbox[ROW_TILE * 4];
  __shared__ __align__(16) float s_tbox[M_TILE * 4];
  __shared__ __align__(16) int   s_tid[M_TILE];

  const int row0 = blockIdx.y * ROW_TILE;
  const int m0   = blockIdx.x * M_TILE;
  const int tid  = threadIdx.x;

#if USE_ASYNC
  // CDNA5 async global->LDS: per-lane DMA into LDS, tracked by ASYNCcnt.
  if (tid < M_TILE) {
    __builtin_amdgcn_global_load_async_to_lds_b128(
        GP128(&tboxes[(size_t)(m0 + tid) * 4]),
        LP128(&s_tbox[tid * 4]), 0, 0);
    __builtin_amdgcn_global_load_async_to_lds_b32(
        GP32(&tids[m0 + tid]),
        LP32(&s_tid[tid]), 0, 0);
  }
  if (tid < ROW_TILE) {
    __builtin_amdgcn_global_load_async_to_lds_b128(
        GP128(&xyxy[(size_t)(row0 + tid) * 4]),
        LP128(&s_rbox[tid * 4]), 0, 0);
  }
  {
    const float* gp = negprob + (size_t)row0 * CP;   // 16*92 floats, contiguous
    for (int c = tid; c < (ROW_TILE * CP) / 4; c += 256) {
      __builtin_amdgcn_global_load_async_to_lds_b128(
          GP128(&gp[c * 4]),
          LP128(&s_prob[c * 4]), 0, 0);
    }
  }
  __builtin_amdgcn_s_wait_asynccnt(0);
  __syncthreads();
#else
  if (tid < M_TILE) {
    *(v4f*)&s_tbox[tid * 4] = *(const v4f*)&tboxes[(size_t)(m0 + tid) * 4];
    s_tid[tid] = tids[m0 + tid];
  }
  if (tid < ROW_TILE)
    *(v4f*)&s_rbox[tid * 4] = *(const v4f*)&xyxy[(size_t)(row0 + tid) * 4];
  {
    const float* gp = negprob + (size_t)row0 * CP;
    for (int c = tid; c < (ROW_TILE * CP) / 4; c += 256)
      *(v4f*)&s_prob[c * 4] = *(const v4f*)&gp[c * 4];
  }
  __syncthreads();
#endif

  const int lane = tid & 31;
  const int wv   = tid >> 5;        // 8 waves -> 8 column subtiles of 16
  const int nloc = lane & 15;       // N (target column) within subtile
  const int hi   = lane >> 4;       // half-wave selector for A/B K layout
  const int msub = wv * 16;

  const int myid = s_tid[msub + nloc];   // one-hot column id for this lane's N

  // cost_class: D = (-P_tile) x onehot over K=0..91 (padded to 92), K=4/step.
  v8f acc = {};
  #pragma unroll
  for (int k0 = 0; k0 < CP; k0 += 4) {
    // A 16x4 f32: lanes 0-15 -> K={k0,k0+1}, lanes 16-31 -> K={k0+2,k0+3}
    v2f a = *(const v2f*)&s_prob[nloc * CP + k0 + hi * 2];
    // B 4x16 one-hot, row K striped across lanes within each VGPR
    v2f b;
    b.x = (myid == k0 + hi)     ? 1.0f : 0.0f;
    b.y = (myid == k0 + 2 + hi) ? 1.0f : 0.0f;
    acc = __builtin_amdgcn_wmma_f32_16x16x4_f32(
        false, a, false, b, (short)0, acc, false, false);
  }

  // Epilogue: L1 + GIoU in VALU, fuse with WMMA class cost, NT-store output.
  const float bx1 = s_tbox[(msub + nloc) * 4 + 0];
  const float by1 = s_tbox[(msub + nloc) * 4 + 1];
  const float bx2 = s_tbox[(msub + nloc) * 4 + 2];
  const float by2 = s_tbox[(msub + nloc) * 4 + 3];
  const float area_b = (bx2 - bx1) * (by2 - by1);
  const int mcol = m0 + msub + nloc;

  #pragma unroll
  for (int j = 0; j < 8; ++j) {
    const int row = j + hi * 8;     // C/D layout: VGPR j, halves -> M = j / j+8
    const float ax1 = s_rbox[row * 4 + 0];
    const float ay1 = s_rbox[row * 4 + 1];
    const float ax2 = s_rbox[row * 4 + 2];
    const float ay2 = s_rbox[row * 4 + 3];

    const float l1 = fabsf(ax1 - bx1) + fabsf(ay1 - by1) +
                     fabsf(ax2 - bx2) + fabsf(ay2 - by2);

    const float area_a = (ax2 - ax1) * (ay2 - ay1);
    const float iw = fmaxf(fminf(ax2, bx2) - fmaxf(ax1, bx1), 0.0f);
    const float ih = fmaxf(fminf(ay2, by2) - fmaxf(ay1, by1), 0.0f);
    const float inter = iw * ih;
    const float uni = area_a + area_b - inter;
    const float iou = inter / uni;
    const float cw = fmaxf(fmaxf(ax2, bx2) - fminf(ax1, bx1), 0.0f);
    const float ch = fmaxf(fmaxf(ay2, by2) - fminf(ay1, by1), 0.0f);
    const float area_c = cw * ch;
    const float giou = iou - (area_c - uni) / area_c;

    const float cost = 5.0f * l1 + acc[j] - 2.0f * giou;
    __builtin_nontemporal_store(cost, &out[(size_t)(row0 + row) * MTGT + mcol]);
  }
}

// ---------------------------------------------------------------------------
extern "C" void kernel_launch(void* const* d_in, const int* in_sizes, int n_in,
                              void* d_out, int out_size, void* d_ws, size_t ws_size,
                              hipStream_t stream) {
  const float* pred_logits = (const float*)d_in[0];  // [16,1000,91]
  const float* pred_boxes  = (const float*)d_in[1];  // [16,1000,4]
  const float* tgt_boxes   = (const float*)d_in[2];  // [2048,4]
  const int*   tgt_ids     = (const int*)d_in[3];    // [2048]
  float* out = (float*)d_out;                        // [16,1000,2048]

  float* negprob = (float*)d_ws;                     // BQ*CP floats (~5.9 MB)
  float* xyxy    = negprob + (size_t)BQ * CP;        // BQ*4 floats (16B aligned)

  // one wave per row, 8 waves per block
  matcher_prep<<<dim3((BQ + 7) / 8), 256, 0, stream>>>(
      pred_logits, pred_boxes, negprob, xyxy);

  matcher_cost<<<dim3(MTGT / M_TILE, BQ / ROW_TILE), 256, 0, stream>>>(
      negprob, xyxy, tgt_boxes, tgt_ids, out);
}